// FlattenModel_62706522521916
// MI455X (gfx1250) — compile-verified
//
#include <hip/hip_runtime.h>
#include <hip/hip_bf16.h>
#include <math.h>

#define K_EDGES 8192
#define N_PTS   1024
#define D_DIM   3
#define TOPK_N  10
#define GT      1024               // greedy kernel threads (32 waves)
#define EPT     (K_EDGES / GT)     // edges per thread in the scan (8)

#ifndef __has_builtin
#define __has_builtin(x) 0
#endif

#if __has_builtin(__builtin_amdgcn_global_load_async_to_lds_b128)
#define HAVE_ASYNC_LDS 1
#else
#define HAVE_ASYNC_LDS 0
#endif

// 16-byte int vector type matching the async-LDS builtin's parameter type.
typedef int v4i_b128 __attribute__((vector_size(16)));
typedef __attribute__((address_space(1))) v4i_b128 gas_v4i;  // global (AS1)
typedef __attribute__((address_space(3))) v4i_b128 las_v4i;  // LDS (AS3)

// ---------------------------------------------------------------------------
// K1: extract per-edge start/end points. s4.w = squared length (for top-k).
// ---------------------------------------------------------------------------
__global__ void extract_endpoints(const float* __restrict__ edges,
                                  float4* __restrict__ s4,
                                  float4* __restrict__ e4) {
  int i = blockIdx.x * blockDim.x + threadIdx.x;
  if (i >= K_EDGES) return;
  const float* row  = edges + (size_t)i * (N_PTS * D_DIM);
  const float* last = row + (size_t)(N_PTS - 1) * D_DIM;
  float sx = row[0],  sy = row[1],  sz = row[2];
  float ex = last[0], ey = last[1], ez = last[2];
  float dx = ex - sx, dy = ey - sy, dz = ez - sz;
  s4[i] = make_float4(sx, sy, sz, dx * dx + dy * dy + dz * dz);
  e4[i] = make_float4(ex, ey, ez, 0.0f);
}

__device__ __forceinline__ float dist2(float cx, float cy, float cz, float4 p) {
  float dx = cx - p.x, dy = cy - p.y, dz = cz - p.z;
  return dx * dx + dy * dy + dz * dz;
}

// ---------------------------------------------------------------------------
// K2: persistent single-workgroup greedy chaining. Entire endpoint set lives
// in LDS (256KB) — CDNA5 320KB LDS makes this possible. Staged via async
// global->LDS. Consumed edges are poisoned with +INF coords (no mask, no
// branch in the hot loop: 2 x ds_load_b128 + FMAs per edge eval).
// ---------------------------------------------------------------------------
__global__ void __launch_bounds__(GT)
greedy_order(const float4* __restrict__ g_s4,
             const float4* __restrict__ g_e4,
             int*   __restrict__ ws_order,
             int*   __restrict__ ws_flips,
             float* __restrict__ out_order,
             float* __restrict__ out_flips,
             float* __restrict__ out_dist) {
  extern __shared__ unsigned char smem[];
  float4* s_s    = (float4*)smem;                 // K float4 (256KB total w/ s_e)
  float4* s_e    = s_s + K_EDGES;
  float*  r_val  = (float*)(s_e + K_EDGES);       // 32
  float*  r_val2 = r_val + 32;                    // 32
  int*    r_idx  = (int*)(r_val2 + 32);           // 32
  float*  s_cur  = (float*)(r_idx + 32);          // 4

  const int   tid  = threadIdx.x;
  const int   lane = tid & 31;
  const int   wave = tid >> 5;
  const float INF  = __builtin_inff();

  // ---- stage endpoints into LDS (async global->LDS, ASYNCcnt tracked) ------
#if HAVE_ASYNC_LDS
  for (int i = tid; i < K_EDGES; i += GT) {
    v4i_b128* gs = (v4i_b128*)const_cast<float4*>(g_s4 + i);
    v4i_b128* ge = (v4i_b128*)const_cast<float4*>(g_e4 + i);
    v4i_b128* ls = (v4i_b128*)(s_s + i);
    v4i_b128* le = (v4i_b128*)(s_e + i);
    __builtin_amdgcn_global_load_async_to_lds_b128((gas_v4i*)gs, (las_v4i*)ls, 0, 0);
    __builtin_amdgcn_global_load_async_to_lds_b128((gas_v4i*)ge, (las_v4i*)le, 0, 0);
  }
#if __has_builtin(__builtin_amdgcn_s_wait_asynccnt)
  __builtin_amdgcn_s_wait_asynccnt(0);
#else
  asm volatile("s_wait_asynccnt 0" ::: "memory");
#endif
#else
  for (int i = tid; i < K_EDGES; i += GT) {
    s_s[i] = g_s4[i];
    s_e[i] = g_e4[i];
  }
#endif
  __syncthreads();

  // ---- top-10 longest edges (stable-descending argsort semantics) ----------
  // Picked candidates are masked by poisoning their .w with -INF.
  int cand[TOPK_N];  // meaningful in thread 0 only
  for (int r = 0; r < TOPK_N; ++r) {
    float bv = -1.0f;
    int   bi = 0x7fffffff;
    for (int e = tid; e < K_EDGES; e += GT) {
      float l2 = s_s[e].w;
      if (l2 > bv || (l2 == bv && e < bi)) { bv = l2; bi = e; }
    }
    for (int off = 16; off > 0; off >>= 1) {
      float ov = __shfl_down(bv, off, 32);
      int   oi = __shfl_down(bi, off, 32);
      if (ov > bv || (ov == bv && oi < bi)) { bv = ov; bi = oi; }
    }
    if (lane == 0) { r_val[wave] = bv; r_idx[wave] = bi; }
    __syncthreads();
    if (wave == 0) {
      float v  = r_val[lane];
      int   ix = r_idx[lane];
      for (int off = 16; off > 0; off >>= 1) {
        float ov = __shfl_down(v, off, 32);
        int   oi = __shfl_down(ix, off, 32);
        if (ov > v || (ov == v && oi < ix)) { v = ov; ix = oi; }
      }
      if (lane == 0) { s_s[ix].w = -INF; r_idx[0] = ix; }
    }
    __syncthreads();
    if (tid == 0) cand[r] = r_idx[0];
  }
  __syncthreads();

  // Reference's score degenerates to -1e-9*cand (its own endpoints give d=0),
  // so start = max index among the candidates.
  if (tid == 0) {
    int si = cand[0];
    for (int r = 1; r < TOPK_N; ++r) si = si > cand[r] ? si : cand[r];
    r_idx[0] = si;
  }
  __syncthreads();
  const int start_index = r_idx[0];

  // ---- choose initial orientation (use_flip0) ------------------------------
  float4 S0 = s_s[start_index];
  float4 E0 = s_e[start_index];
  float ms = INF, me = INF;
  for (int e = tid; e < K_EDGES; e += GT) {
    if (e == start_index) continue;
    float4 a = s_s[e], b = s_e[e];
    ms = fminf(ms, fminf(dist2(S0.x, S0.y, S0.z, a), dist2(S0.x, S0.y, S0.z, b)));
    me = fminf(me, fminf(dist2(E0.x, E0.y, E0.z, a), dist2(E0.x, E0.y, E0.z, b)));
  }
  for (int off = 16; off > 0; off >>= 1) {
    ms = fminf(ms, __shfl_down(ms, off, 32));
    me = fminf(me, __shfl_down(me, off, 32));
  }
  if (lane == 0) { r_val[wave] = ms; r_val2[wave] = me; }
  __syncthreads();

  float fpx = 0.f, fpy = 0.f, fpz = 0.f;  // first point (thread 0 registers)
  if (wave == 0) {
    float v0 = r_val[lane], v1 = r_val2[lane];
    for (int off = 16; off > 0; off >>= 1) {
      v0 = fminf(v0, __shfl_down(v0, off, 32));
      v1 = fminf(v1, __shfl_down(v1, off, 32));
    }
    if (lane == 0) {
      bool flip0 = v0 < v1;                 // min-to-others(s0) < min-to-others(e0)
      float4 ce = flip0 ? S0 : E0;          // current chain end
      float4 fp = flip0 ? E0 : S0;          // first point of reordered[0]
      s_cur[0] = ce.x; s_cur[1] = ce.y; s_cur[2] = ce.z;
      fpx = fp.x; fpy = fp.y; fpz = fp.z;
      // consume the start edge: poison its endpoints (all scans are done —
      // we are past the __syncthreads that follows the flip0 scan)
      float4 poison = make_float4(INF, INF, INF, INF);
      s_s[start_index] = poison;
      s_e[start_index] = poison;
      ws_order[0]  = start_index;
      ws_flips[0]  = flip0 ? 1 : 0;
      out_order[0] = (float)start_index;
      out_flips[0] = flip0 ? 1.0f : 0.0f;
    }
  }
  __syncthreads();

  // ---- 8191 sequential greedy argmin steps, all data in LDS ----------------
  for (int step = 1; step < K_EDGES; ++step) {
    float cx = s_cur[0], cy = s_cur[1], cz = s_cur[2];
    float bv = INF;
    int   bi = 0x7fffffff;
#pragma unroll
    for (int it = 0; it < EPT; ++it) {
      int   e = tid + it * GT;
      float da2 = dist2(cx, cy, cz, s_s[e]);   // poisoned edges -> +INF
      float db2 = dist2(cx, cy, cz, s_e[e]);
      float m   = fminf(da2, db2);
      if (m < bv || (m == bv && e < bi)) { bv = m; bi = e; }
    }
    for (int off = 16; off > 0; off >>= 1) {
      float ov = __shfl_down(bv, off, 32);
      int   oi = __shfl_down(bi, off, 32);
      if (ov < bv || (ov == bv && oi < bi)) { bv = ov; bi = oi; }
    }
    if (lane == 0) { r_val[wave] = bv; r_idx[wave] = bi; }
    __syncthreads();
    if (wave == 0) {
      float v  = r_val[lane];
      int   ix = r_idx[lane];
      for (int off = 16; off > 0; off >>= 1) {
        float ov = __shfl_down(v, off, 32);
        int   oi = __shfl_down(ix, off, 32);
        if (ov < v || (ov == v && oi < ix)) { v = ov; ix = oi; }
      }
      if (lane == 0) {
        int    j   = ix;
        float4 a   = s_s[j];
        float4 b   = s_e[j];
        float  da2 = dist2(cx, cy, cz, a);
        float  db2 = dist2(cx, cy, cz, b);
        bool   flip = da2 > db2;
        float4 ne   = flip ? a : b;
        float4 poison = make_float4(INF, INF, INF, INF);
        s_s[j] = poison;                      // consume edge j
        s_e[j] = poison;
        s_cur[0] = ne.x; s_cur[1] = ne.y; s_cur[2] = ne.z;
        ws_order[step]     = j;
        ws_flips[step]     = flip ? 1 : 0;
        out_order[step]    = (float)j;
        out_flips[step]    = flip ? 1.0f : 0.0f;
        out_dist[step - 1] = sqrtf(v);
      }
    }
    __syncthreads();
  }

  if (tid == 0) {
    float dx = s_cur[0] - fpx, dy = s_cur[1] - fpy, dz = s_cur[2] - fpz;
    out_dist[K_EDGES - 1] = sqrtf(dx * dx + dy * dy + dz * dz);  // d_close
  }
}

// ---------------------------------------------------------------------------
// K3: bandwidth phase — gather rows by order, reverse if flipped. ~200MB
// through HBM: ~8.6us at 23.3 TB/s roofline.
// ---------------------------------------------------------------------------
__global__ void gather_flip(const float* __restrict__ edges,
                            const int* __restrict__ ws_order,
                            const int* __restrict__ ws_flips,
                            float* __restrict__ out) {
  int b = blockIdx.x;
  int j = ws_order[b];
  int f = ws_flips[b];
  const float* src = edges + (size_t)j * (N_PTS * D_DIM);
  float*       dst = out   + (size_t)b * (N_PTS * D_DIM);
  if (!f) {
    const float4* s4 = (const float4*)src;            // row = 12KB, 16B aligned
    float4*       d4 = (float4*)dst;
    for (int i = threadIdx.x; i < (N_PTS * D_DIM) / 4; i += blockDim.x)
      d4[i] = s4[i];
  } else {
    for (int p = threadIdx.x; p < N_PTS; p += blockDim.x) {
      int q = N_PTS - 1 - p;
      dst[p * 3 + 0] = src[q * 3 + 0];
      dst[p * 3 + 1] = src[q * 3 + 1];
      dst[p * 3 + 2] = src[q * 3 + 2];
    }
  }
}

// ---------------------------------------------------------------------------
extern "C" void kernel_launch(void* const* d_in, const int* in_sizes, int n_in,
                              void* d_out, int out_size, void* d_ws, size_t ws_size,
                              hipStream_t stream) {
  const float* edges = (const float*)d_in[0];

  float* out           = (float*)d_out;
  float* out_reordered = out;                                     // K*N*D
  float* out_order     = out + (size_t)K_EDGES * N_PTS * D_DIM;   // K
  float* out_flips     = out_order + K_EDGES;                     // K
  float* out_dist      = out_flips + K_EDGES;                     // K

  unsigned char* ws = (unsigned char*)d_ws;
  float4* g_s4     = (float4*)ws;
  float4* g_e4     = g_s4 + K_EDGES;
  int*    ws_order = (int*)(g_e4 + K_EDGES);
  int*    ws_flips = ws_order + K_EDGES;

  extract_endpoints<<<K_EDGES / 256, 256, 0, stream>>>(edges, g_s4, g_e4);

  size_t smem = (size_t)K_EDGES * 2 * sizeof(float4)  // endpoint set: 256KB
              + 2 * 32 * sizeof(float)                // reduction vals
              + 32 * sizeof(int)                      // reduction idx
              + 4 * sizeof(float);                    // current chain end
  greedy_order<<<1, GT, smem, stream>>>(g_s4, g_e4, ws_order, ws_flips,
                                        out_order, out_flips, out_dist);

  gather_flip<<<K_EDGES, 256, 0, stream>>>(edges, ws_order, ws_flips, out_reordered);
}